// RoiPooling_73607149518869
// MI455X (gfx1250) — compile-verified
//
#include <hip/hip_runtime.h>
#include <hip/hip_bf16.h>

typedef float __attribute__((ext_vector_type(2))) v2f;
typedef float __attribute__((ext_vector_type(4))) v4f;
typedef float __attribute__((ext_vector_type(8))) v8f;

#define CCH 512
#define HH  256
#define WW  256
#define NROI 4096
#define NBOX 16384           // patch_num(4) * NROI
#define HWC ((size_t)HH * WW * CCH)
#define LDSTR 260            // padded LDS row stride (floats): 16B aligned, bank-shift 4

struct BoxInfo { int x0, x1, y0, y1; float area; int mask; };

// ---------------------------------------------------------------- box corners
__global__ __launch_bounds__(256) void box_kernel(const float* __restrict__ roi,
                                                  BoxInfo* __restrict__ bi) {
    int b = blockIdx.x * blockDim.x + threadIdx.x;
    if (b >= NBOX) return;
    int p = b >> 12;                 // patch index (outer), N = 4096
    int n = b & (NROI - 1);
    float xmin = roi[n * 4 + 0], ymin = roi[n * 4 + 1];
    float xmax = roi[n * 4 + 2], ymax = roi[n * 4 + 3];
    int ix = p & 1, iy = p >> 1;     // pe = 2
    float ws = (xmax - xmin) * 0.5f;
    float hs = (ymax - ymin) * 0.5f;
    float ax = xmin + (float)ix * ws;
    float ay = ymin + (float)iy * hs;
    auto clampi = [](float v) {
        float r = rintf(v);                      // round-half-to-even == jnp.round
        r = fminf(fmaxf(r, 0.0f), 255.0f);
        return (int)r;
    };
    int x0 = clampi(ax), x1 = clampi(ax + ws);
    int y0 = clampi(ay), y1 = clampi(ay + hs);
    int cw = x1 - x0, ch = y1 - y0;
    int m  = (cw >= 1 && ch >= 1) ? 1 : 0;
    int ar = cw * ch; if (ar < 1) ar = 1;
    BoxInfo out; out.x0 = x0; out.x1 = x1; out.y0 = y0; out.y1 = y1;
    out.area = (float)ar; out.mask = m;
    bi[b] = out;
}

// ------------------------------------------- WMMA row (x) prefix scan, fp32
// One wave handles a group = (h, 16 channels). Y = L * X as 16x16x4 WMMA:
// A = constant triangular/ones chunks, B = data (k = x, n = channel),
// D tiles (m = x, n = channel) stored channel-contiguous into S(H,W,C).
__global__ __launch_bounds__(64) void row_scan_wmma(const float* __restrict__ fm,
                                                    float* __restrict__ St) {
    __shared__ float lds[2][16 * LDSTR];
    const int wave = threadIdx.x >> 5;
    const int lane = threadIdx.x & 31;
    const int group = blockIdx.x * 2 + wave;   // 8192 groups
    const int cb = group & 31;                 // channel block (16 ch)
    const int h  = group >> 5;                 // row
    const int c0 = cb * 16;
    const int half = lane >> 4;
    const int col  = lane & 15;

    // stage 16 rows (one per channel) of 256 floats, coalesced float4 loads
    float* myLds = lds[wave];
#pragma unroll
    for (int r = 0; r < 16; ++r) {
        const float* src = fm + (size_t)(c0 + r) * (HH * WW) + (size_t)h * WW;
        float* dst = myLds + r * LDSTR;
        *(v4f*)(dst + lane * 4)       = *(const v4f*)(src + lane * 4);
        *(v4f*)(dst + 128 + lane * 4) = *(const v4f*)(src + 128 + lane * 4);
    }
    __syncthreads();

    // A-matrix constants per ISA layout: lane(half,col): .x=A[m=col,k=2*half], .y=A[m=col,k=2*half+1]
    v2f a_ones; a_ones.x = 1.0f; a_ones.y = 1.0f;
    v2f a_tri[4];
#pragma unroll
    for (int u = 0; u < 4; ++u) {
        a_tri[u].x = ((4 * u + 2 * half + 0) <= col) ? 1.0f : 0.0f;
        a_tri[u].y = ((4 * u + 2 * half + 1) <= col) ? 1.0f : 0.0f;
    }

    const size_t outRow = (size_t)h * WW;      // in (H,W,C): ((h*W + x)*C + c)
#pragma unroll
    for (int J = 0; J < 16; ++J) {
        v8f acc = {};
        for (int kb = 0; kb < 4 * J; ++kb) {   // full ones blocks (k < 16J)
            v2f b = *(const v2f*)(myLds + col * LDSTR + 4 * kb + 2 * half);
            acc = __builtin_amdgcn_wmma_f32_16x16x4_f32(
                false, a_ones, false, b, (short)0, acc, false, false);
        }
#pragma unroll
        for (int u = 0; u < 4; ++u) {          // diagonal (triangular) blocks
            v2f b = *(const v2f*)(myLds + col * LDSTR + 4 * (4 * J + u) + 2 * half);
            acc = __builtin_amdgcn_wmma_f32_16x16x4_f32(
                false, a_tri[u], false, b, (short)0, acc, false, false);
        }
        // store D: VGPR i -> x = 16J + i + 8*half, lanes sweep 16 contiguous channels
#pragma unroll
        for (int i = 0; i < 8; ++i) {
            int x = 16 * J + i + 8 * half;
            St[(outRow + x) * CCH + c0 + col] = acc[i];
        }
    }
}

// ---------------------------------- column (y) prefix scan in place on S(H,W,C)
__global__ __launch_bounds__(256) void col_scan(float* __restrict__ St) {
    size_t off = ((size_t)blockIdx.x * blockDim.x + threadIdx.x) * 4; // over W*C = 131072 floats
    float* p = St + off;
    v4f run = {};
    for (int y = 0; y < HH; ++y) {
        v4f v = *(v4f*)p;
        run += v;
        *(v4f*)p = run;
        p += (size_t)WW * CCH;
    }
}

// ------------------------------------------------ 4-corner gather + normalize
__global__ __launch_bounds__(128) void gather_kernel(const float* __restrict__ St,
                                                     const BoxInfo* __restrict__ bi,
                                                     float* __restrict__ out,
                                                     int* __restrict__ maskOut) {
    int b = blockIdx.x;
    BoxInfo box = bi[b];
    int t = threadIdx.x;
    int c = t * 4;
    int X1 = box.x1 - 1, X0 = box.x0 - 1;
    int Y1 = box.y1 - 1, Y0 = box.y0 - 1;
    v4f s = {};
    if (Y1 >= 0 && X1 >= 0) s += *(const v4f*)(St + ((size_t)(Y1 * WW + X1)) * CCH + c);
    if (Y0 >= 0 && X1 >= 0) s -= *(const v4f*)(St + ((size_t)(Y0 * WW + X1)) * CCH + c);
    if (Y1 >= 0 && X0 >= 0) s -= *(const v4f*)(St + ((size_t)(Y1 * WW + X0)) * CCH + c);
    if (Y0 >= 0 && X0 >= 0) s += *(const v4f*)(St + ((size_t)(Y0 * WW + X0)) * CCH + c);
    v4f r;
    r.x = s.x / box.area; r.y = s.y / box.area;
    r.z = s.z / box.area; r.w = s.w / box.area;
    if (!box.mask) { r.x = 0.f; r.y = 0.f; r.z = 0.f; r.w = 0.f; }
    *(v4f*)(out + (size_t)b * CCH + c) = r;
    if (t == 0) maskOut[b] = box.mask;
}

// ------------------------------ fallback: direct region sums (no integral img)
__global__ __launch_bounds__(256) void direct_kernel(const float* __restrict__ fm,
                                                     const BoxInfo* __restrict__ bi,
                                                     float* __restrict__ out,
                                                     int* __restrict__ maskOut) {
    int b = blockIdx.x;
    BoxInfo box = bi[b];
    int lane = threadIdx.x & 31;
    int cg   = threadIdx.x >> 5;           // 8 channel groups
    for (int c = cg; c < CCH; c += 8) {
        float acc = 0.0f;
        const float* pc = fm + (size_t)c * (HH * WW);
        for (int y = box.y0; y < box.y1; ++y) {
            const float* py = pc + (size_t)y * WW;
            for (int x = box.x0 + lane; x < box.x1; x += 32) acc += py[x];
        }
        for (int o = 16; o > 0; o >>= 1) acc += __shfl_down(acc, o, 32);
        if (lane == 0)
            out[(size_t)b * CCH + c] = box.mask ? (acc / box.area) : 0.0f;
    }
    if (threadIdx.x == 0) maskOut[b] = box.mask;
}

extern "C" void kernel_launch(void* const* d_in, const int* in_sizes, int n_in,
                              void* d_out, int out_size, void* d_ws, size_t ws_size,
                              hipStream_t stream) {
    (void)in_sizes; (void)n_in; (void)out_size;
    const float* fm  = (const float*)d_in[0];
    const float* roi = (const float*)d_in[1];
    float* out = (float*)d_out;
    int* maskOut = (int*)d_out + (size_t)NBOX * CCH;  // mask int32 appended after pooled

    const size_t S_BYTES = HWC * sizeof(float);       // 134 MB integral image
    char* wsb = (char*)d_ws;

    if (ws_size >= S_BYTES + NBOX * sizeof(BoxInfo)) {
        float*   St = (float*)wsb;
        BoxInfo* bi = (BoxInfo*)(wsb + S_BYTES);
        box_kernel   <<<NBOX / 256, 256, 0, stream>>>(roi, bi);
        row_scan_wmma<<<4096, 64, 0, stream>>>(fm, St);     // 8192 groups, 2 waves/blk
        col_scan     <<<(WW * CCH / 4) / 256, 256, 0, stream>>>(St);
        gather_kernel<<<NBOX, 128, 0, stream>>>(St, bi, out, maskOut);
    } else {
        BoxInfo* bi = (BoxInfo*)wsb;
        box_kernel   <<<NBOX / 256, 256, 0, stream>>>(roi, bi);
        direct_kernel<<<NBOX, 256, 0, stream>>>(fm, bi, out, maskOut);
    }
}